// PrototypeLayer_26018911879738
// MI455X (gfx1250) — compile-verified
//
#include <hip/hip_runtime.h>
#include <stdint.h>

typedef float v2f __attribute__((ext_vector_type(2)));
typedef float v8f __attribute__((ext_vector_type(8)));

#define D 64
#define M 512
#define WAVES 8
#define ROWS_PER_BLOCK 128   /* 8 waves * 16 rows */
#define BIGVAL 1000000.0f

// ---- monotonic float<->uint key mapping (order-preserving, works for +/-) ----
__device__ __forceinline__ unsigned fkey(float f) {
    unsigned b = __float_as_uint(f);
    return (b & 0x80000000u) ? ~b : (b | 0x80000000u);
}
__device__ __forceinline__ float funkey(unsigned k) {
    return __uint_as_float((k & 0x80000000u) ? (k ^ 0x80000000u) : ~k);
}

// ---------------------------------------------------------------------------
// K0: prototype squared norms + init column-min keys
// ---------------------------------------------------------------------------
__global__ void pp_init_kernel(const float* __restrict__ proto,
                               float* __restrict__ ws_pp,
                               unsigned* __restrict__ ws_colkey) {
    int j = threadIdx.x;  // 512 threads
    const float4* p = (const float4*)(proto + j * D);
    float s = 0.f;
#pragma unroll
    for (int i = 0; i < D / 4; ++i) {
        float4 v = p[i];
        s += v.x * v.x + v.y * v.y + v.z * v.z + v.w * v.w;
    }
    ws_pp[j] = s;
    ws_colkey[j] = 0xFFFFFFFFu;
}

// ---------------------------------------------------------------------------
// K1: main fused kernel
//   - x copy to out
//   - d_fp = xx + pp - 2 x.p  via V_WMMA_F32_16X16X4_F32, stored to out
//   - per-row min -> per-block partial sums (deterministic)
//   - per-column min -> global atomicMin on ordered keys (order-independent)
// Block: 256 threads (8 waves); each wave owns a 16-row strip, loops all 32
// column tiles with A fragments resident in VGPRs; B served from swizzled LDS.
// ---------------------------------------------------------------------------
__global__ void __launch_bounds__(256)
proto_main_kernel(const float* __restrict__ x,
                  const float* __restrict__ proto,
                  const float* __restrict__ ws_pp,
                  unsigned* __restrict__ ws_colkey,
                  float* __restrict__ ws_blocksum,
                  float* __restrict__ out_x,
                  float* __restrict__ out_laten) {
    __shared__ __align__(16) float protoLds[M * D];   // 128 KB, XOR-swizzled K
    __shared__ unsigned colminLds[M];                 // ordered-key col mins
    __shared__ float waveRowSum[WAVES];

    const int tid  = threadIdx.x;
    const int lane = tid & 31;
    const int wave = tid >> 5;
    const int lo   = lane & 15;   // column-ish index
    const int hi   = lane >> 4;   // K-half / row-half select
    const float INF = __uint_as_float(0x7F800000u);

    // ---- stage prototypes into LDS with bank-conflict-free swizzle ----
    // element (col,k) stored at col*64 + (k ^ (2*(col&15)) ^ (32*(col&1)))
#pragma unroll
    for (int i = 0; i < (M * D) / (256 * 4); ++i) {   // 32 float4 per thread
        int f   = (i * 256 + tid) * 4;                // flat idx, multiple of 4
        int col = f >> 6;
        int k   = f & 63;
        int e   = (2 * (col & 15)) ^ (32 * (col & 1));
        float4 v = *(const float4*)(proto + f);
        *(float2*)&protoLds[col * D + ((k)     ^ e)] = make_float2(v.x, v.y);
        *(float2*)&protoLds[col * D + ((k + 2) ^ e)] = make_float2(v.z, v.w);
    }
    if (tid < M) colminLds[tid] = 0xFFFFFFFFu;
    __syncthreads();

    const int rowBase = blockIdx.x * ROWS_PER_BLOCK + wave * 16;

    // ---- forward pass copy: out_x = x (coalesced float4) ----
    {
        const float* src = x + (size_t)rowBase * D;
        float*       dst = out_x + (size_t)rowBase * D;
#pragma unroll
        for (int i = 0; i < 8; ++i) {
            int idx = i * 128 + lane * 4;
            *(float4*)(dst + idx) = *(const float4*)(src + idx);
        }
    }

    // ---- load A fragments: lane holds row (rowBase+lo), K pair (s*4+2*hi) ----
    v2f afrag[16];
    {
        const float* arow = x + (size_t)(rowBase + lo) * D + hi * 2;
#pragma unroll
        for (int s = 0; s < 16; ++s) {
            float2 v = *(const float2*)(arow + s * 4);
            afrag[s].x = v.x;
            afrag[s].y = v.y;
        }
    }

    // ---- xx = ||x_row||^2 (halves combined across lane pairs) ----
    float xx = 0.f;
#pragma unroll
    for (int s = 0; s < 16; ++s) xx += afrag[s].x * afrag[s].x + afrag[s].y * afrag[s].y;
    xx += __shfl_xor(xx, 16, 32);

    // fold -2 into A (exact scaling by power of two)
#pragma unroll
    for (int s = 0; s < 16; ++s) { afrag[s].x *= -2.f; afrag[s].y *= -2.f; }

    // broadcast xx to the 8 rows this lane's C registers cover
    float xxv[8];
#pragma unroll
    for (int r = 0; r < 8; ++r) xxv[r] = __shfl(xx, r + 8 * hi, 32);

    float vmin[8];
#pragma unroll
    for (int r = 0; r < 8; ++r) vmin[r] = INF;

    // ---- loop over 32 column tiles of 16 prototypes ----
    const int ecol = (2 * lo) ^ (32 * (lo & 1));  // read-side swizzle (col&15==lo, col&1==lo&1)
    for (int t = 0; t < M / 16; ++t) {
        const int col = t * 16 + lo;
        const float* bcol = &protoLds[col * D];

        v8f acc = {};
#pragma unroll
        for (int s = 0; s < 16; ++s) {
            int koff = s * 4 + hi * 2;
            float2 bv = *(const float2*)&bcol[koff ^ ecol];
            v2f b; b.x = bv.x; b.y = bv.y;
            // D = A*B + C   (A pre-scaled by -2)  -> acc = -2 * x . p
            acc = __builtin_amdgcn_wmma_f32_16x16x4_f32(
                false, afrag[s], false, b, (short)0, acc, false, false);
        }

        const float pp = ws_pp[col];
        float cmin = INF;
#pragma unroll
        for (int r = 0; r < 8; ++r) {
            float dval = (xxv[r] + pp) + acc[r];
            out_laten[(size_t)(rowBase + r + 8 * hi) * M + col] = dval;
            vmin[r] = fminf(vmin[r], dval);
            cmin = fminf(cmin, dval);
        }
        // column min over this wave's 16 rows
        cmin = fminf(cmin, __shfl_xor(cmin, 16, 32));
        if (hi == 0) atomicMin(&colminLds[col], fkey(cmin));
    }

    // ---- row mins: reduce across the 16 column-lanes ----
#pragma unroll
    for (int r = 0; r < 8; ++r) {
        vmin[r] = fminf(vmin[r], __shfl_xor(vmin[r], 1, 32));
        vmin[r] = fminf(vmin[r], __shfl_xor(vmin[r], 2, 32));
        vmin[r] = fminf(vmin[r], __shfl_xor(vmin[r], 4, 32));
        vmin[r] = fminf(vmin[r], __shfl_xor(vmin[r], 8, 32));
    }
    // lane group lo-half now holds mins of rows 0..7, hi-half rows 8..15
    float s8 = 0.f;
#pragma unroll
    for (int r = 0; r < 8; ++r) s8 += vmin[r];
    s8 += __shfl_xor(s8, 16, 32);
    if (lane == 0) waveRowSum[wave] = s8;
    __syncthreads();

    if (tid == 0) {
        float bs = 0.f;
#pragma unroll
        for (int w = 0; w < WAVES; ++w) bs += waveRowSum[w];
        ws_blocksum[blockIdx.x] = bs;   // deterministic per-block partial
    }
    if (tid < M) atomicMin(&ws_colkey[tid], colminLds[tid]);  // order-independent
}

// ---------------------------------------------------------------------------
// K2: prototype-to-prototype min distances (R_3), zero self-dist masked to BIG.
// Identical summation order for ||a||^2, ||b||^2 and a.b guarantees the
// self-distance is exactly 0 and therefore masked, matching the reference.
// ---------------------------------------------------------------------------
__global__ void p2p_kernel(const float* __restrict__ proto,
                           float* __restrict__ ws_p2pmin) {
    __shared__ float pj[D];
    __shared__ float red[128];
    const int j = blockIdx.x;
    const int t = threadIdx.x;  // 128 threads
    if (t < D) pj[t] = proto[j * D + t];
    __syncthreads();

    float ppa = 0.f;
#pragma unroll
    for (int k = 0; k < D; ++k) ppa += pj[k] * pj[k];

    float best = __uint_as_float(0x7F800000u);
    for (int jj = t; jj < M; jj += 128) {
        const float* pb = proto + jj * D;
        float dot = 0.f, ppb = 0.f;
#pragma unroll
        for (int k = 0; k < D; ++k) {
            dot += pj[k] * pb[k];
            ppb += pb[k] * pb[k];
        }
        float dv = (ppa + ppb) - 2.f * dot;
        if (dv == 0.f) dv = BIGVAL;
        best = fminf(best, dv);
    }
    red[t] = best;
    __syncthreads();
    for (int s = 64; s > 0; s >>= 1) {
        if (t < s) red[t] = fminf(red[t], red[t + s]);
        __syncthreads();
    }
    if (t == 0) ws_p2pmin[j] = red[0];
}

// ---------------------------------------------------------------------------
// K3: deterministic final reductions -> R_1, R_2, R_3 scalars
// ---------------------------------------------------------------------------
__global__ void finalize_kernel(const float* __restrict__ ws_blocksum,
                                const unsigned* __restrict__ ws_colkey,
                                const float* __restrict__ ws_p2pmin,
                                float* __restrict__ out_scalars,
                                int n, int nblocks) {
    __shared__ float red[512];
    const int t = threadIdx.x;  // 512 threads

    // R_2 = mean over rows of row-min
    float s = 0.f;
    for (int b = t; b < nblocks; b += 512) s += ws_blocksum[b];
    red[t] = s; __syncthreads();
    for (int k = 256; k > 0; k >>= 1) { if (t < k) red[t] += red[t + k]; __syncthreads(); }
    float R2 = red[0] / (float)n;
    __syncthreads();

    // R_1 = mean over prototypes of column-min
    red[t] = funkey(ws_colkey[t]); __syncthreads();
    for (int k = 256; k > 0; k >>= 1) { if (t < k) red[t] += red[t + k]; __syncthreads(); }
    float R1 = red[0] / (float)M;
    __syncthreads();

    // R_3 = mean over prototypes of masked p2p min
    red[t] = ws_p2pmin[t]; __syncthreads();
    for (int k = 256; k > 0; k >>= 1) { if (t < k) red[t] += red[t + k]; __syncthreads(); }
    if (t == 0) {
        out_scalars[0] = R1;
        out_scalars[1] = R2;
        out_scalars[2] = red[0] / (float)M;
    }
}

// ---------------------------------------------------------------------------
extern "C" void kernel_launch(void* const* d_in, const int* in_sizes, int n_in,
                              void* d_out, int out_size, void* d_ws, size_t ws_size,
                              hipStream_t stream) {
    const float* x     = (const float*)d_in[0];
    const float* proto = (const float*)d_in[1];
    const int n = in_sizes[0] / D;          // 131072
    const int nblocks = n / ROWS_PER_BLOCK; // 1024

    float* out = (float*)d_out;
    const size_t nd = (size_t)n * D;
    float* out_x       = out;           // x passthrough
    float* out_scalars = out + nd;      // R_1, R_2, R_3
    float* out_laten   = out + nd + 3;  // [n, 512] laten_similarity

    float*    ws_pp       = (float*)d_ws;                 // 512 f32
    unsigned* ws_colkey   = (unsigned*)(ws_pp + 512);     // 512 u32
    float*    ws_blocksum = ws_pp + 1024;                 // nblocks f32
    float*    ws_p2pmin   = ws_pp + 1024 + 2048;          // 512 f32

    pp_init_kernel<<<1, 512, 0, stream>>>(proto, ws_pp, ws_colkey);
    proto_main_kernel<<<nblocks, 256, 0, stream>>>(x, proto, ws_pp, ws_colkey,
                                                   ws_blocksum, out_x, out_laten);
    p2p_kernel<<<M, 128, 0, stream>>>(proto, ws_p2pmin);
    finalize_kernel<<<1, 512, 0, stream>>>(ws_blocksum, ws_colkey, ws_p2pmin,
                                           out_scalars, n, nblocks);
}